// ParamDecoder_80625126080848
// MI455X (gfx1250) — compile-verified
//
#include <hip/hip_runtime.h>
#include <math.h>

// ---------------------------------------------------------------------------
// MI455X (gfx1250, wave32) implementation.
// All matrix math uses V_WMMA_F32_16X16X4_F32 (exact fp32 on the matrix pipe).
// A-tiles are staged into LDS by the Tensor Data Mover (tensor_load_to_lds,
// TENSORcnt) with hardware padding to the conflict-free LDS stride; B-tiles
// are stored k-pair-interleaved so every WMMA B fragment is one ds_load_b64.
//
// WMMA f32 16x16x4 per-lane layout (ISA 7.12.2):
//  A (16x4):  lane L holds M=L%16;  a.x=A[M][2*(L/16)],   a.y=A[M][2*(L/16)+1]
//  B (4x16):  lane L holds N=L%16;  b.x=B[2*(L/16)][N],   b.y=B[2*(L/16)+1][N]
//  C/D: VGPR r: lanes0-15 -> (M=r, N=lane); lanes16-31 -> (M=r+8, N=lane-16)
// ---------------------------------------------------------------------------

typedef float v2f __attribute__((ext_vector_type(2)));
typedef float v8f __attribute__((ext_vector_type(8)));
typedef unsigned int u32x4 __attribute__((ext_vector_type(4)));
typedef int i32x4 __attribute__((ext_vector_type(4)));
typedef int i32x8 __attribute__((ext_vector_type(8)));

__device__ __forceinline__ v8f wmma_f32(v2f a, v2f b, v8f c) {
  return __builtin_amdgcn_wmma_f32_16x16x4_f32(false, a, false, b, (short)0, c,
                                               false, false);
}

// ---------------------------------------------------------------------------
// TDM: async 2D fp32 tile load Global -> LDS, with LDS padding:
// pad_interval = 16 DWORDs (code 3), pad_amount = 2 DWORDs (code 1)
// -> LDS row stride = 16 + 2 = 18 floats (matches AS_LD).
// tensor_w/tensor_h are the remaining tensor extent from the tile origin, so
// out-of-bounds rows/cols are zero-filled by the TDM engine.
// ---------------------------------------------------------------------------
__device__ __forceinline__ void tdm_load_tile_f32(unsigned lds_off,
                                                  const float* gaddr, int rows,
                                                  int cols, int tensor_w,
                                                  int tensor_h,
                                                  int row_stride) {
  unsigned long long ga = (unsigned long long)(uintptr_t)gaddr;
  u32x4 g0;
  g0[0] = 1u;  // count=1 (valid user descriptor)
  g0[1] = lds_off;
  g0[2] = (unsigned)(ga & 0xffffffffu);
  g0[3] = (unsigned)((ga >> 32) & 0x01ffffffu) | (2u << 30);  // type=2 (image)
  i32x8 g1;
  // data_size=2 (4B) | pad_enable | pad_interval=3 (16 dw) | pad_amount=1 (2 dw)
  g1[0] = (2 << 16) | (1 << 20) | (3 << 22) | (1 << 25);
  g1[1] = (tensor_w & 0xffff) << 16;  // tensor_dim0[15:0] -> bits 63:48
  g1[2] = ((unsigned)tensor_w >> 16) | ((tensor_h & 0xffff) << 16);
  g1[3] = ((unsigned)tensor_h >> 16) | ((cols & 0xffff) << 16);  // tile_dim0
  g1[4] = rows & 0xffff;  // tile_dim1 (tile_dim2 = 0)
  g1[5] = row_stride;     // tensor_dim0_stride[31:0]
  g1[6] = 0;
  g1[7] = 0;
  i32x4 z = {};
#if __has_include(<hip/amd_detail/amd_gfx1250_TDM.h>)
  i32x8 z8 = {};
  __builtin_amdgcn_tensor_load_to_lds(g0, g1, z, z, z8, 0);  // clang-23 form
#else
  __builtin_amdgcn_tensor_load_to_lds(g0, g1, z, z, 0);      // ROCm 7.2 form
#endif
}

// ---------------------------------------------------------------------------
// Generic GEMM:  D[M,N] = act(A[M,K] @ W + bias)
//   transW==0: W is (K,N) row-major, row stride ldW
//   transW==1: W is (N,K) row-major, row stride ldW   (i.e. x @ W^T)
// 128 threads = 4 waves. 64x64 output tile, K staged in 16-wide LDS chunks.
// A tile arrives via TDM (wave 0 issues, TENSORcnt); W tile is staged by all
// threads k-pair-interleaved so B fragments are single aligned ds_load_b64.
// ---------------------------------------------------------------------------
#define AS_LD 18  // even (8B-aligned pairs) and 18%64 -> conflict-free

__global__ __launch_bounds__(128) void k_gemm(
    const float* __restrict__ A, const float* __restrict__ W,
    const float* __restrict__ bias, float* __restrict__ D, int M, int N, int K,
    int ldA, int ldW, int ldD, int transW, int relu) {
  __shared__ float As[64 * AS_LD];
  __shared__ float Ws[16 * 64];  // interleaved: [(k>>1)*64+n]*2 + (k&1)

  const int tid = threadIdx.x;
  const int wave = tid >> 5;
  const int lane = tid & 31;
  const int ln = lane & 15;
  const int l16 = lane >> 4;
  const int tileM = blockIdx.y * 64;
  const int tileN = blockIdx.x * 64;
  const unsigned asOff = (unsigned)(uintptr_t)&As[0];

  v8f acc[4] = {};

  for (int k0 = 0; k0 < K; k0 += 16) {
    // async TDM stage of the 64x16 A tile (OOB -> zero, pad -> stride 18)
    if (wave == 0) {
      tdm_load_tile_f32(asOff, A + (size_t)tileM * ldA + k0, 64, 16, K - k0,
                        M - tileM, ldA);
    }
    // stage W tile, k-pair interleaved
    for (int idx = tid; idx < 16 * 64; idx += 128) {
      int k = idx >> 6, n = idx & 63;
      int gk = k0 + k, gn = tileN + n;
      float v = 0.0f;
      if (gk < K && gn < N)
        v = transW ? W[(size_t)gn * ldW + gk] : W[(size_t)gk * ldW + gn];
      Ws[((((k >> 1) << 6) + n) << 1) | (k & 1)] = v;
    }
    // prefetch next W K-chunk into the cache hierarchy (global_prefetch)
    if (k0 + 16 < K) {
      if (!transW) {
        if (tid < 16 && k0 + 16 + tid < K)
          __builtin_prefetch(&W[(size_t)(k0 + 16 + tid) * ldW + tileN], 0, 1);
      } else {
        if (tid < 64 && tileN + tid < N)
          __builtin_prefetch(&W[(size_t)(tileN + tid) * ldW + k0 + 16], 0, 1);
      }
    }
    if (wave == 0) __builtin_amdgcn_s_wait_tensorcnt(0);
    __syncthreads();

#pragma unroll
    for (int kk = 0; kk < 16; kk += 4) {
      const int ka = kk + 2 * l16;
      v2f a = *(const v2f*)&As[(wave * 16 + ln) * AS_LD + ka];
#pragma unroll
      for (int nt = 0; nt < 4; ++nt) {
        v2f b = *(const v2f*)&Ws[((((ka >> 1) << 6) + nt * 16 + ln) << 1)];
        acc[nt] = wmma_f32(a, b, acc[nt]);
      }
    }
    __syncthreads();
  }

#pragma unroll
  for (int nt = 0; nt < 4; ++nt) {
#pragma unroll
    for (int r = 0; r < 8; ++r) {
      int row = tileM + wave * 16 + r + 8 * l16;
      int col = tileN + nt * 16 + ln;
      if (row < M && col < N) {
        float v = acc[nt][r];
        if (bias) v += bias[col];
        if (relu) v = fmaxf(v, 0.0f);
        D[(size_t)row * ldD + col] = v;
      }
    }
  }
}

// ---------------------------------------------------------------------------
// LayerNorm over 512 features, optional residual:  Y = LN(X (+R)) * g + b
// ---------------------------------------------------------------------------
__global__ __launch_bounds__(256) void k_ln(const float* __restrict__ X,
                                            const float* __restrict__ R,
                                            const float* __restrict__ g,
                                            const float* __restrict__ b,
                                            float* __restrict__ Y) {
  __shared__ float red[256];
  const int row = blockIdx.x;
  const int t = threadIdx.x;
  const float* xr = X + (size_t)row * 512;
  float x0 = xr[t], x1 = xr[t + 256];
  if (R) {
    const float* rr = R + (size_t)row * 512;
    x0 += rr[t];
    x1 += rr[t + 256];
  }
  red[t] = x0 + x1;
  __syncthreads();
  for (int o = 128; o > 0; o >>= 1) {
    if (t < o) red[t] += red[t + o];
    __syncthreads();
  }
  const float mean = red[0] * (1.0f / 512.0f);
  __syncthreads();
  const float d0 = x0 - mean, d1 = x1 - mean;
  red[t] = d0 * d0 + d1 * d1;
  __syncthreads();
  for (int o = 128; o > 0; o >>= 1) {
    if (t < o) red[t] += red[t + o];
    __syncthreads();
  }
  const float rs = rsqrtf(red[0] * (1.0f / 512.0f) + 1e-5f);
  Y[(size_t)row * 512 + t] = d0 * rs * g[t] + b[t];
  Y[(size_t)row * 512 + t + 256] = d1 * rs * g[t + 256] + b[t + 256];
}

// ---------------------------------------------------------------------------
// Fused MHA core: one workgroup (256 threads = 8 waves) per (batch, head).
// Q,K,V layout: (B, S, NH, dh) flattened == (B*S, 512) with head stride 64.
// K,V tiles + 64x256 score block live in dynamic LDS (~197KB of 320KB WGP LDS).
// V is stored s-pair-interleaved -> P@V B fragments are single ds_load_b64.
// ---------------------------------------------------------------------------
#define KS_LD 68   // even, 68%64=4 -> conflict-free, 8B-aligned pairs
#define SB_LD 260  // even, 260%64=4 -> conflict-free, 8B-aligned pairs

__global__ __launch_bounds__(256) void k_attn(const float* __restrict__ Q,
                                              const float* __restrict__ Kk,
                                              const float* __restrict__ V,
                                              float* __restrict__ O) {
  extern __shared__ float lds[];
  float* Ks = lds;                  // 256 * KS_LD
  float* Vs = Ks + 256 * KS_LD;     // 256 * 64, interleaved
  float* Sb = Vs + 256 * 64;        // 64 * SB_LD

  const int bh = blockIdx.x;
  const int b = bh >> 3;
  const int h = bh & 7;
  const int tid = threadIdx.x;
  const int wave = tid >> 5;
  const int lane = tid & 31;
  const int ln = lane & 15;
  const int l16 = lane >> 4;
  const size_t hoff = (size_t)h * 64;

  for (int idx = tid; idx < 256 * 64; idx += 256) {
    int s = idx >> 6, k = idx & 63;
    float kv = Kk[((size_t)(b * 256 + s)) * 512 + hoff + k];
    float vv = V[((size_t)(b * 256 + s)) * 512 + hoff + k];
    Ks[s * KS_LD + k] = kv;
    Vs[((((s >> 1) << 6) + k) << 1) | (s & 1)] = vv;
  }
  __syncthreads();

  for (int qb = 0; qb < 4; ++qb) {
    const int qbase = qb * 64;
    // ---- S = (Q @ K^T) / 8 : 64x256
    {
      const int rt = wave & 3;
      const int ctg = (wave >> 2) * 8;
      v8f sacc[8] = {};
      const int m = qbase + rt * 16 + ln;
      const float* qrow = Q + ((size_t)(b * 256 + m)) * 512 + hoff;
#pragma unroll
      for (int k0 = 0; k0 < 64; k0 += 4) {
        const int ka = k0 + 2 * l16;
        v2f a = *(const v2f*)(qrow + ka);
#pragma unroll
        for (int i = 0; i < 8; ++i) {
          int n = (ctg + i) * 16 + ln;
          v2f bb = *(const v2f*)&Ks[n * KS_LD + ka];
          sacc[i] = wmma_f32(a, bb, sacc[i]);
        }
      }
#pragma unroll
      for (int i = 0; i < 8; ++i)
#pragma unroll
        for (int r = 0; r < 8; ++r) {
          int row = rt * 16 + r + 8 * l16;
          int col = (ctg + i) * 16 + ln;
          Sb[row * SB_LD + col] = sacc[i][r] * 0.125f;  // 1/sqrt(64)
        }
    }
    __syncthreads();
    // ---- softmax rows (threads 0..63 each own a row)
    if (tid < 64) {
      float* sr = Sb + tid * SB_LD;
      float mx = -3.4e38f;
      for (int j = 0; j < 256; ++j) mx = fmaxf(mx, sr[j]);
      float sum = 0.0f;
      for (int j = 0; j < 256; ++j) {
        float e = __expf(sr[j] - mx);
        sr[j] = e;
        sum += e;
      }
      float inv = 1.0f / sum;
      for (int j = 0; j < 256; ++j) sr[j] *= inv;
    }
    __syncthreads();
    // ---- O = P @ V : 64x64
    {
      const int rt = wave >> 1;
      const int ctb = (wave & 1) * 2;
      v8f oacc[2] = {};
      const int m = rt * 16 + ln;
      for (int k0 = 0; k0 < 256; k0 += 4) {
        const int ka = k0 + 2 * l16;
        v2f a = *(const v2f*)&Sb[m * SB_LD + ka];
#pragma unroll
        for (int i = 0; i < 2; ++i) {
          int n = (ctb + i) * 16 + ln;
          v2f bb = *(const v2f*)&Vs[((((ka >> 1) << 6) + n) << 1)];
          oacc[i] = wmma_f32(a, bb, oacc[i]);
        }
      }
#pragma unroll
      for (int i = 0; i < 2; ++i)
#pragma unroll
        for (int r = 0; r < 8; ++r) {
          int row = qbase + rt * 16 + r + 8 * l16;
          int col = (ctb + i) * 16 + ln;
          O[((size_t)(b * 256 + row)) * 512 + hoff + col] = oacc[i][r];
        }
    }
    __syncthreads();
  }
}

// ---------------------------------------------------------------------------
// Persistent GRU scan. One 1024-thread workgroup (32 waves) loops timesteps.
// h (64x512) lives in LDS; gh (64x1536) round-trips through L2 scratch.
// ---------------------------------------------------------------------------
#define HS_LD 516  // even, 516%64=4 -> conflict-free, 8B-aligned pairs

__global__ __launch_bounds__(1024) void k_gru_scan(
    const float* __restrict__ GI, const float* __restrict__ Whh,
    const float* __restrict__ bhh, float* __restrict__ gh_ws,
    float* __restrict__ Yout, int outLd, int outOff, int T, int reverse,
    const float* __restrict__ wih_pos,  // = w_ih + 512 (stride 513) or nullptr
    const float* __restrict__ head_w, const float* __restrict__ head_b,
    float* __restrict__ pos_out) {
  extern __shared__ float lds[];
  float* hS = lds;                 // 64 * HS_LD
  float* posS = hS + 64 * HS_LD;   // 64

  const int tid = threadIdx.x;
  const int wave = tid >> 5;
  const int lane = tid & 31;
  const int ln = lane & 15;
  const int l16 = lane >> 4;
  const int rt = wave >> 3;  // row tile 0..3
  const int cg = wave & 7;   // col group, 12 col tiles each

  for (int idx = tid; idx < 64 * HS_LD; idx += 1024) hS[idx] = 0.0f;
  if (tid < 64) posS[tid] = 0.0f;
  __syncthreads();

  for (int tstep = 0; tstep < T; ++tstep) {
    const int tt = reverse ? (T - 1 - tstep) : tstep;

    // Phase A: gh = h @ Whh^T  (64x1536, K=512), two passes of 6 tiles/wave
    for (int pass = 0; pass < 2; ++pass) {
      v8f acc[6] = {};
      const int ctb = cg * 12 + pass * 6;
      const int m = rt * 16 + ln;
      for (int k0 = 0; k0 < 512; k0 += 4) {
        const int ka = k0 + 2 * l16;
        v2f a = *(const v2f*)&hS[m * HS_LD + ka];
#pragma unroll
        for (int i = 0; i < 6; ++i) {
          v2f bb =
              *(const v2f*)(Whh + (size_t)((ctb + i) * 16 + ln) * 512 + ka);
          acc[i] = wmma_f32(a, bb, acc[i]);
        }
      }
#pragma unroll
      for (int i = 0; i < 6; ++i)
#pragma unroll
        for (int r = 0; r < 8; ++r) {
          int row = rt * 16 + r + 8 * l16;
          int col = (ctb + i) * 16 + ln;
          gh_ws[(size_t)row * 1536 + col] = acc[i][r];
        }
    }
    __threadfence_block();
    __syncthreads();

    // Phase B: elementwise gates, update h (and outputs)
    for (int idx = tid; idx < 64 * 512; idx += 1024) {
      const int bb_ = idx >> 9;
      const int j = idx & 511;
      const float* gi = GI + (size_t)(bb_ * T + tt) * 1536;
      float gir = gi[j], giz = gi[j + 512], gin = gi[j + 1024];
      if (wih_pos) {
        float p = posS[bb_];
        gir += p * wih_pos[(size_t)j * 513];
        giz += p * wih_pos[(size_t)(j + 512) * 513];
        gin += p * wih_pos[(size_t)(j + 1024) * 513];
      }
      float ghr = gh_ws[(size_t)bb_ * 1536 + j] + bhh[j];
      float ghz = gh_ws[(size_t)bb_ * 1536 + j + 512] + bhh[j + 512];
      float ghn = gh_ws[(size_t)bb_ * 1536 + j + 1024] + bhh[j + 1024];
      float rg = 1.0f / (1.0f + __expf(-(gir + ghr)));
      float zg = 1.0f / (1.0f + __expf(-(giz + ghz)));
      float ng = tanhf(gin + rg * ghn);
      float hn = (1.0f - zg) * ng + zg * hS[bb_ * HS_LD + j];
      hS[bb_ * HS_LD + j] = hn;
      if (Yout) Yout[(size_t)(bb_ * T + tt) * outLd + outOff + j] = hn;
    }
    __syncthreads();

    if (head_w) {  // pgru: pos += h @ head_w + head_b ; emit pos
      if (tid < 64) {
        float d = 0.0f;
        for (int j = 0; j < 512; ++j) d += hS[tid * HS_LD + j] * head_w[j];
        float p = posS[tid] + d + head_b[0];
        posS[tid] = p;
        pos_out[tid * T + tt] = p;
      }
      __syncthreads();
    }
  }
}

// tgt[b,s,:] = query_embed[s,:] + cf[b,:] + tf[b,s,:]
__global__ __launch_bounds__(256) void k_build_tgt(const float* __restrict__ qe,
                                                   const float* __restrict__ cf,
                                                   const float* __restrict__ tf,
                                                   float* __restrict__ tgt) {
  size_t idx = (size_t)blockIdx.x * 256 + threadIdx.x;
  int j = (int)(idx & 511);
  int s = (int)((idx >> 9) & 255);
  int b = (int)(idx >> 17);
  tgt[idx] = qe[s * 512 + j] + cf[b * 512 + j] + tf[idx];
}

// ---------------------------------------------------------------------------
extern "C" void kernel_launch(void* const* d_in, const int* in_sizes, int n_in,
                              void* d_out, int out_size, void* d_ws,
                              size_t ws_size, hipStream_t stream) {
  (void)in_sizes; (void)n_in; (void)out_size; (void)ws_size;
  const int M = 64 * 256;  // 16384 token rows

  auto P = [&](int i) { return (const float*)d_in[i]; };
  enum {
    IN_SRC = 0, IN_RECV = 1, IN_TX = 2,
    P_QE = 3, P_CW1 = 4, P_CB1 = 5, P_CW2 = 6, P_CB2 = 7,
    P_TW1 = 8, P_TB1 = 9, P_TW2 = 10, P_TB2 = 11,
    P_PROJW = 12, P_PROJB = 13,
    ENC0 = 14,             // 3 layers x 16 tensors
    ENC_NG = 62, ENC_NB = 63,
    DEC0 = 64,             // 3 layers x 26 tensors
    DEC_NG = 142, DEC_NB = 143,
    PG_WIH = 144, PG_WHH = 145, PG_BIH = 146, PG_BHH = 147,
    HEAD_W = 148, HEAD_B = 149,
    FR0 = 150,             // 4 GRUs x 4 tensors
    FREQ_W = 166, FREQ_B = 167
  };

  float* ws = (float*)d_ws;
  const size_t BUF = (size_t)M * 512;
  float* Aw = ws;            // also GI base (A..C = 16384x1536)
  float* Bw = ws + BUF;
  float* Cw = ws + 2 * BUF;  // FFN hidden spans C..D (16384x1024)
  float* Dw = ws + 3 * BUF;  // XC1 spans D..E
  float* Ew = ws + 4 * BUF;
  float* Fw = ws + 5 * BUF;  // XC2 spans F..G
  float* Gw = ws + 6 * BUF;
  float* Hw = ws + 7 * BUF;  // encoder memory
  float* ghS = ws + 8 * BUF;            // 64x1536
  float* cfS = ghS + 64 * 1536;         // 64x512
  float* chS = cfS + 64 * 512;          // 64x512

  auto gemm = [&](const float* A, int ldA, const float* W, int ldW, int transW,
                  const float* bias, float* D, int ldD, int m, int n, int k,
                  int relu) {
    dim3 grid((n + 63) / 64, (m + 63) / 64);
    k_gemm<<<grid, 128, 0, stream>>>(A, W, bias, D, m, n, k, ldA, ldW, ldD,
                                     transW, relu);
  };
  auto ln = [&](const float* X, const float* R, const float* g, const float* b,
                float* Y) { k_ln<<<M, 256, 0, stream>>>(X, R, g, b, Y); };
  const size_t attnLds = (size_t)(256 * KS_LD + 256 * 64 + 64 * SB_LD) * 4;
  auto attn = [&](const float* q, const float* k, const float* v, float* o) {
    k_attn<<<512, 256, attnLds, stream>>>(q, k, v, o);
  };
  const size_t gruLds = (size_t)(64 * HS_LD + 64) * 4;

  // 1) src projection -> A
  gemm(P(IN_SRC), 768, P(P_PROJW), 512, 0, P(P_PROJB), Aw, 512, M, 512, 768, 0);
  // 2) coord MLP -> cf
  gemm(P(IN_RECV), 2, P(P_CW1), 512, 0, P(P_CB1), chS, 512, 64, 512, 2, 1);
  gemm(chS, 512, P(P_CW2), 512, 0, P(P_CB2), cfS, 512, 64, 512, 512, 0);
  // 3) tx MLP -> C
  gemm(P(IN_TX), 1, P(P_TW1), 512, 0, P(P_TB1), Bw, 512, M, 512, 1, 1);
  gemm(Bw, 512, P(P_TW2), 512, 0, P(P_TB2), Cw, 512, M, 512, 512, 0);
  // 4) tgt -> G
  k_build_tgt<<<(unsigned)(((size_t)M * 512) / 256), 256, 0, stream>>>(
      P(P_QE), cfS, Cw, Gw);

  // 5) encoder (stream in A)
  for (int i = 0; i < 3; ++i) {
    int bs = ENC0 + 16 * i;
    gemm(Aw, 512, P(bs + 0), 512, 0, P(bs + 4), Cw, 512, M, 512, 512, 0);
    gemm(Aw, 512, P(bs + 1), 512, 0, P(bs + 5), Dw, 512, M, 512, 512, 0);
    gemm(Aw, 512, P(bs + 2), 512, 0, P(bs + 6), Ew, 512, M, 512, 512, 0);
    attn(Cw, Dw, Ew, Fw);
    gemm(Fw, 512, P(bs + 3), 512, 0, P(bs + 7), Cw, 512, M, 512, 512, 0);
    ln(Aw, Cw, P(bs + 12), P(bs + 13), Bw);
    gemm(Bw, 512, P(bs + 8), 1024, 0, P(bs + 9), Cw, 1024, M, 1024, 512, 1);
    gemm(Cw, 1024, P(bs + 10), 512, 0, P(bs + 11), Ew, 512, M, 512, 1024, 0);
    ln(Bw, Ew, P(bs + 14), P(bs + 15), Aw);
  }
  ln(Aw, nullptr, P(ENC_NG), P(ENC_NB), Hw);  // memory -> H

  // 6) decoder (stream in G, memory H)
  for (int i = 0; i < 3; ++i) {
    int bs = DEC0 + 26 * i;
    gemm(Gw, 512, P(bs + 0), 512, 0, P(bs + 4), Cw, 512, M, 512, 512, 0);
    gemm(Gw, 512, P(bs + 1), 512, 0, P(bs + 5), Dw, 512, M, 512, 512, 0);
    gemm(Gw, 512, P(bs + 2), 512, 0, P(bs + 6), Ew, 512, M, 512, 512, 0);
    attn(Cw, Dw, Ew, Fw);
    gemm(Fw, 512, P(bs + 3), 512, 0, P(bs + 7), Cw, 512, M, 512, 512, 0);
    ln(Gw, Cw, P(bs + 20), P(bs + 21), Bw);
    gemm(Bw, 512, P(bs + 8), 512, 0, P(bs + 12), Cw, 512, M, 512, 512, 0);
    gemm(Hw, 512, P(bs + 9), 512, 0, P(bs + 13), Dw, 512, M, 512, 512, 0);
    gemm(Hw, 512, P(bs + 10), 512, 0, P(bs + 14), Ew, 512, M, 512, 512, 0);
    attn(Cw, Dw, Ew, Fw);
    gemm(Fw, 512, P(bs + 11), 512, 0, P(bs + 15), Dw, 512, M, 512, 512, 0);
    ln(Bw, Dw, P(bs + 22), P(bs + 23), Aw);
    gemm(Aw, 512, P(bs + 16), 1024, 0, P(bs + 17), Cw, 1024, M, 1024, 512, 1);
    gemm(Cw, 1024, P(bs + 18), 512, 0, P(bs + 19), Ew, 512, M, 512, 1024, 0);
    ln(Aw, Ew, P(bs + 24), P(bs + 25), Gw);
  }
  ln(Gw, nullptr, P(DEC_NG), P(DEC_NB), Fw);  // decoder out -> F

  float* freq_out = (float*)d_out;                 // (B,256,34)
  float* pos_out = freq_out + (size_t)M * 34;      // (B,256,1)

  // 7) pgru: gi for the o_t part (w_ih[:, :512]^T), then persistent scan
  gemm(Fw, 512, P(PG_WIH), 513, 1, P(PG_BIH), Aw, 1536, M, 1536, 512, 0);
  k_gru_scan<<<1, 1024, gruLds, stream>>>(
      Aw, P(PG_WHH), P(PG_BHH), ghS, nullptr, 0, 0, 256, 0,
      P(PG_WIH) + 512, P(HEAD_W), P(HEAD_B), pos_out);

  // 8) frnn layer 1 (input F, concat into XC1 = D..E, ld 1024)
  for (int d = 0; d < 2; ++d) {
    int g = FR0 + 4 * d;
    gemm(Fw, 512, P(g + 0), 512, 1, P(g + 2), Aw, 1536, M, 1536, 512, 0);
    k_gru_scan<<<1, 1024, gruLds, stream>>>(Aw, P(g + 1), P(g + 3), ghS, Dw,
                                            1024, d * 512, 256, d, nullptr,
                                            nullptr, nullptr, nullptr);
  }
  // 9) frnn layer 2 (input XC1, concat into XC2 = F..G)
  for (int d = 0; d < 2; ++d) {
    int g = FR0 + 4 * (2 + d);
    gemm(Dw, 1024, P(g + 0), 1024, 1, P(g + 2), Aw, 1536, M, 1536, 1024, 0);
    k_gru_scan<<<1, 1024, gruLds, stream>>>(Aw, P(g + 1), P(g + 3), ghS, Fw,
                                            1024, d * 512, 256, d, nullptr,
                                            nullptr, nullptr, nullptr);
  }
  // 10) freq head -> d_out
  gemm(Fw, 1024, P(FREQ_W), 34, 0, P(FREQ_B), freq_out, 34, M, 34, 1024, 0);
}